// GCN_12867722019435
// MI455X (gfx1250) — compile-verified
//
#include <hip/hip_runtime.h>
#include <hip/hip_bf16.h>

typedef float v2f __attribute__((ext_vector_type(2)));
typedef float v8f __attribute__((ext_vector_type(8)));

// ---------------------------------------------------------------------------
// Dense GEMM C[M,N] = op(A)[M,K] @ W[K,N] using V_WMMA_F32_16X16X4_F32.
// One wave32 computes one 16x16 tile of C. Grid: x = M/16 tiles, y covers
// N-tiles in groups of 8 (8 waves / 256-thread block).
// RELU template param applies max(a,0) to A elements (fuses layer-1 ReLU
// into layer-2's GEMM read).
//
// VGPR layouts per CDNA5 ISA 7.12.2:
//   A (16x4 f32):  lanes 0-15 -> M=lane, v0=K0,v1=K1 ; lanes 16-31 -> K2,K3
//   B (4x16 f32):  lanes 0-15 -> N=lane, v0=K0,v1=K1 ; lanes 16-31 -> K2,K3
//   C (16x16 f32): VGPR g: lanes 0-15 -> M=g, lanes 16-31 -> M=8+g; N=lane&15
// ---------------------------------------------------------------------------
template <int RELU>
__global__ __launch_bounds__(256) void gcn_gemm_wmma_f32(
    const float* __restrict__ A, const float* __restrict__ W,
    float* __restrict__ C, int M, int N, int K)
{
    const int wave = threadIdx.x >> 5;
    const int lane = threadIdx.x & 31;
    const int half = lane >> 4;       // 0: K pair {0,1}, 1: K pair {2,3}
    const int l15  = lane & 15;

    const int ntile = blockIdx.y * 8 + wave;
    if (ntile * 16 >= N) return;      // wave-uniform exit; EXEC stays all-1s
    const int m0 = blockIdx.x * 16;
    const int n0 = ntile * 16;

    const float* arow = A + (size_t)(m0 + l15) * K + 2 * half;       // row of A
    const float* wcol = W + (size_t)(2 * half) * N + (n0 + l15);     // col of W

    v8f acc = {};
    #pragma unroll 4
    for (int k = 0; k < K; k += 4) {
        v2f a, b;
        a.x = arow[k];
        a.y = arow[k + 1];
        if (RELU) { a.x = fmaxf(a.x, 0.0f); a.y = fmaxf(a.y, 0.0f); }
        b.x = wcol[(size_t)k * N];
        b.y = wcol[(size_t)(k + 1) * N];
        // (neg_a, A, neg_b, B, c_mod, C, reuse_a, reuse_b)
        acc = __builtin_amdgcn_wmma_f32_16x16x4_f32(
            false, a, false, b, (short)0, acc, false, false);
    }

    float* crow = C + (size_t)(m0 + 8 * half) * N + (n0 + l15);
    #pragma unroll
    for (int g = 0; g < 8; ++g)
        crow[(size_t)g * N] = acc[g];
}

// ---------------------------------------------------------------------------
// Edge scatter-add: out[dst] += S[src] for every edge, rows of `feat` floats.
// One thread handles a 4-float chunk of one edge's row (float4 gather +
// 4 hardware f32 atomic adds, which resolve in L2 on MI455X).
// featShift: log2(feat) (8 for 256, 7 for 128).
// ---------------------------------------------------------------------------
__global__ __launch_bounds__(256) void gcn_scatter_add(
    const float* __restrict__ S,
    const int* __restrict__ esrc, const int* __restrict__ edst,
    float* __restrict__ out, int nEdges, int featShift)
{
    const int chunkShift = featShift - 2;
    long long idx = (long long)blockIdx.x * blockDim.x + threadIdx.x;
    long long total = (long long)nEdges << chunkShift;
    if (idx >= total) return;

    const int e = (int)(idx >> chunkShift);
    const int c = (int)(idx & ((1 << chunkShift) - 1));
    const int s = esrc[e];
    const int d = edst[e];

    const float4 v = ((const float4*)(S + ((size_t)s << featShift)))[c];
    float* dst = out + ((size_t)d << featShift) + ((size_t)c << 2);
    unsafeAtomicAdd(dst + 0, v.x);
    unsafeAtomicAdd(dst + 1, v.y);
    unsafeAtomicAdd(dst + 2, v.z);
    unsafeAtomicAdd(dst + 3, v.w);
}

extern "C" void kernel_launch(void* const* d_in, const int* in_sizes, int n_in,
                              void* d_out, int out_size, void* d_ws, size_t ws_size,
                              hipStream_t stream) {
    const float* x    = (const float*)d_in[0];   // [50000, 256]
    const float* w1   = (const float*)d_in[1];   // [256, 256]
    const float* w2   = (const float*)d_in[2];   // [256, 128]
    const int*   esrc = (const int*)d_in[3];     // [800000]
    const int*   edst = (const int*)d_in[4];     // [800000]
    float*       out  = (float*)d_out;           // [50000, 128]

    const int NN   = 50000;          // nodes (50000 = 3125 * 16, exact tiles)
    const int NF   = 256;            // input features
    const int NH   = 256;            // hidden
    const int NO   = 128;            // output
    const int nE   = in_sizes[3];    // edges

    // Workspace layout: [support: 50000*256 f32][agg1: 50000*256 f32]
    float* support = (float*)d_ws;                       // reused for layer-2 support
    float* agg1    = support + (size_t)NN * NF;

    dim3 blk(256);

    // --- Layer 1 GEMM: support = x @ w1  (N=256 -> 16 n-tiles -> grid.y = 2)
    gcn_gemm_wmma_f32<0><<<dim3(NN / 16, NH / 16 / 8), blk, 0, stream>>>(
        x, w1, support, NN, NH, NF);

    // --- Layer 1 aggregate: agg1 = segment_sum(support[src] -> dst)
    hipMemsetAsync(agg1, 0, (size_t)NN * NH * sizeof(float), stream);
    {
        long long tot = (long long)nE * (NH / 4);
        gcn_scatter_add<<<dim3((unsigned)((tot + 255) / 256)), blk, 0, stream>>>(
            support, esrc, edst, agg1, nE, 8);
    }

    // --- Layer 2 GEMM: support = relu(agg1) @ w2  (N=128 -> 8 n-tiles -> grid.y = 1)
    gcn_gemm_wmma_f32<1><<<dim3(NN / 16, NO / 16 / 8), blk, 0, stream>>>(
        agg1, w2, support, NN, NO, NH);

    // --- Layer 2 aggregate: out = segment_sum(support[src] -> dst)
    hipMemsetAsync(out, 0, (size_t)out_size * sizeof(float), stream);
    {
        long long tot = (long long)nE * (NO / 4);
        gcn_scatter_add<<<dim3((unsigned)((tot + 255) / 256)), blk, 0, stream>>>(
            support, esrc, edst, out, nE, 7);
    }
}